// Nonlocal_62294205661400
// MI455X (gfx1250) — compile-verified
//
#include <hip/hip_runtime.h>

typedef float v2f __attribute__((ext_vector_type(2)));
typedef float v8f __attribute__((ext_vector_type(8)));

#define NB   2
#define NC   64
#define NCI  32
#define NPIX 9216   // 96*96

__device__ __forceinline__ v8f wmma_f32(v2f a, v2f b, v8f c) {
    // D = A(16x4 f32) * B(4x16 f32) + C  -> v_wmma_f32_16x16x4_f32
    return __builtin_amdgcn_wmma_f32_16x16x4_f32(
        /*neg_a=*/false, a, /*neg_b=*/false, b,
        /*c_mod=*/(short)0, c, /*reuse_a=*/false, /*reuse_b=*/false);
}

// ---------------------------------------------------------------------------
// K0: zero the S accumulator (atomics accumulate into it)
// ---------------------------------------------------------------------------
__global__ void zero_kernel(float* __restrict__ s, int n) {
    int i = blockIdx.x * blockDim.x + threadIdx.x;
    if (i < n) s[i] = 0.0f;
}

// ---------------------------------------------------------------------------
// K1: fused projections. One wave owns one (batch, 16-col) strip and computes
// all 3 projections x 2 row-tiles (6 16x16 tiles). X tile loaded ONCE per
// k-step and reused by 6 WMMAs.
// ---------------------------------------------------------------------------
__global__ __launch_bounds__(256) void proj_kernel(
    const float* __restrict__ x,
    const float* __restrict__ tw, const float* __restrict__ tb,
    const float* __restrict__ pw, const float* __restrict__ pb,
    const float* __restrict__ gw, const float* __restrict__ gb,
    float* __restrict__ Pt, float* __restrict__ Pp, float* __restrict__ Pg) {
    const int CT = NPIX / 16;                       // 576 column tiles
    int wid  = blockIdx.x * 8 + (threadIdx.x >> 5); // global wave id
    int lane = threadIdx.x & 31;

    int ct = wid % CT;
    int b  = wid / CT;                              // batch

    const float* X = x + (size_t)b * NC * NPIX;
    int m    = lane & 15;
    int half = lane >> 4;                           // 0: K{0,1}, 1: K{2,3}
    int col  = ct * 16 + m;

    const float* W[3]    = {tw, pw, gw};
    const float* bias[3] = {tb, pb, gb};
    float*       out[3]  = {Pt, Pp, Pg};

    v8f acc[3][2] = {};
    for (int k0 = 0; k0 < NC; k0 += 4) {
        v2f bm;
        bm.x = X[(size_t)(k0 + half * 2) * NPIX + col];
        bm.y = X[(size_t)(k0 + half * 2 + 1) * NPIX + col];
        #pragma unroll
        for (int p = 0; p < 3; ++p) {
            #pragma unroll
            for (int rt = 0; rt < 2; ++rt) {
                v2f a;
                int mrow = rt * 16 + m;
                a.x = W[p][mrow * NC + k0 + half * 2];
                a.y = W[p][mrow * NC + k0 + half * 2 + 1];
                acc[p][rt] = wmma_f32(a, bm, acc[p][rt]);
            }
        }
    }
    #pragma unroll
    for (int p = 0; p < 3; ++p) {
        float* O = out[p] + (size_t)b * NCI * NPIX;
        #pragma unroll
        for (int rt = 0; rt < 2; ++rt) {
            #pragma unroll
            for (int v = 0; v < 8; ++v) {
                int row = rt * 16 + v + half * 8;
                O[(size_t)row * NPIX + col] = acc[p][rt][v] + bias[p][row];
            }
        }
    }
}

// ---------------------------------------------------------------------------
// K2: S(32x32) = P_phi(32xN) * P_g(32xN)^T per batch. One wave owns one
// (batch, 128-col chunk) and computes all four 16x16 S tiles, reusing each
// A/B fragment twice. fp32 atomic accumulation into zeroed S.
// ---------------------------------------------------------------------------
__global__ __launch_bounds__(256) void smat_kernel(
    const float* __restrict__ Pp, const float* __restrict__ Pg,
    float* __restrict__ S) {
    const int NSPLIT = 72;                          // 9216 / 128
    int wid  = blockIdx.x * 8 + (threadIdx.x >> 5);
    int lane = threadIdx.x & 31;

    int chunk = wid % NSPLIT;
    int b     = wid / NSPLIT;

    int m    = lane & 15;
    int half = lane >> 4;
    const float* Ab = Pp + (size_t)b * NCI * NPIX;
    const float* Bb = Pg + (size_t)b * NCI * NPIX;

    int n0 = chunk * 128;
    v8f acc[2][2] = {};
    for (int n = n0; n < n0 + 128; n += 4) {
        v2f a[2], bb[2];
        #pragma unroll
        for (int t = 0; t < 2; ++t) {
            const float* Ar = Ab + (size_t)(t * 16 + m) * NPIX;
            const float* Br = Bb + (size_t)(t * 16 + m) * NPIX;
            a[t].x  = Ar[n + half * 2];  a[t].y  = Ar[n + half * 2 + 1];
            bb[t].x = Br[n + half * 2];  bb[t].y = Br[n + half * 2 + 1];
        }
        #pragma unroll
        for (int it = 0; it < 2; ++it)
            #pragma unroll
            for (int jt = 0; jt < 2; ++jt)
                acc[it][jt] = wmma_f32(a[it], bb[jt], acc[it][jt]);
    }
    float* Sb = S + b * NCI * NCI;
    #pragma unroll
    for (int it = 0; it < 2; ++it)
        #pragma unroll
        for (int jt = 0; jt < 2; ++jt)
            #pragma unroll
            for (int v = 0; v < 8; ++v) {
                int row = it * 16 + v + half * 8;
                atomicAdd(&Sb[row * NCI + (jt * 16 + m)], acc[it][jt][v]);
            }
}

// ---------------------------------------------------------------------------
// K3: fold  M(b,64,32) = rec_w(64x32) * S(b)^T / N   (tiny, plain VALU)
// ---------------------------------------------------------------------------
__global__ void fold_kernel(const float* __restrict__ S,
                            const float* __restrict__ rw,
                            float* __restrict__ M64) {
    int i = blockIdx.x * blockDim.x + threadIdx.x;  // NB*64*32 = 4096
    if (i >= NB * NC * NCI) return;
    int k = i % NCI;
    int c = (i / NCI) % NC;
    int b = i / (NCI * NC);
    const float* Sb = S + b * NCI * NCI;
    float acc = 0.0f;
    #pragma unroll
    for (int j = 0; j < NCI; ++j)
        acc += rw[c * NCI + j] * Sb[k * NCI + j];
    M64[i] = acc * (1.0f / (float)NPIX);
}

// ---------------------------------------------------------------------------
// K4: Z(64xN) = X + M(64x32) * P_theta(32xN) + rec_b. One wave owns one
// (batch, 16-col) strip: all 4 row-tiles, P_theta tile loaded once per k-step.
// ---------------------------------------------------------------------------
__global__ __launch_bounds__(256) void out_kernel(
    const float* __restrict__ x, const float* __restrict__ Pt,
    const float* __restrict__ M64, const float* __restrict__ rb,
    float* __restrict__ z) {
    const int CT = NPIX / 16;
    int wid  = blockIdx.x * 8 + (threadIdx.x >> 5);
    int lane = threadIdx.x & 31;

    int ct = wid % CT;
    int b  = wid / CT;

    int m    = lane & 15;
    int half = lane >> 4;
    int col  = ct * 16 + m;
    const float* Mb = M64 + b * NC * NCI;
    const float* Bm = Pt + (size_t)b * NCI * NPIX;

    v8f acc[4] = {};
    #pragma unroll
    for (int k0 = 0; k0 < NCI; k0 += 4) {
        v2f bb;
        bb.x = Bm[(size_t)(k0 + half * 2) * NPIX + col];
        bb.y = Bm[(size_t)(k0 + half * 2 + 1) * NPIX + col];
        #pragma unroll
        for (int rt = 0; rt < 4; ++rt) {
            v2f a;
            const float* Ar = Mb + (rt * 16 + m) * NCI;
            a.x = Ar[k0 + half * 2];
            a.y = Ar[k0 + half * 2 + 1];
            acc[rt] = wmma_f32(a, bb, acc[rt]);
        }
    }
    const float* X = x + (size_t)b * NC * NPIX;
    float*       Z = z + (size_t)b * NC * NPIX;
    #pragma unroll
    for (int rt = 0; rt < 4; ++rt)
        #pragma unroll
        for (int v = 0; v < 8; ++v) {
            int c = rt * 16 + v + half * 8;
            Z[(size_t)c * NPIX + col] = acc[rt][v] + X[(size_t)c * NPIX + col] + rb[c];
        }
}

// ---------------------------------------------------------------------------
extern "C" void kernel_launch(void* const* d_in, const int* in_sizes, int n_in,
                              void* d_out, int out_size, void* d_ws, size_t ws_size,
                              hipStream_t stream) {
    const float* x  = (const float*)d_in[0];
    const float* tw = (const float*)d_in[1];
    const float* tb = (const float*)d_in[2];
    const float* pw = (const float*)d_in[3];
    const float* pb = (const float*)d_in[4];
    const float* gw = (const float*)d_in[5];
    const float* gb = (const float*)d_in[6];
    const float* rw = (const float*)d_in[7];
    const float* rb = (const float*)d_in[8];

    float* ws  = (float*)d_ws;
    float* Pt  = ws;                                   // B*32*N
    float* Pp  = Pt + (size_t)NB * NCI * NPIX;         // B*32*N
    float* Pg  = Pp + (size_t)NB * NCI * NPIX;         // B*32*N
    float* S   = Pg + (size_t)NB * NCI * NPIX;         // B*32*32
    float* M64 = S + NB * NCI * NCI;                   // B*64*32

    zero_kernel<<<(NB * NCI * NCI + 255) / 256, 256, 0, stream>>>(S, NB * NCI * NCI);

    // NB * 576 col strips = 1152 waves / 8 per block
    proj_kernel<<<(NB * (NPIX / 16)) / 8, 256, 0, stream>>>(
        x, tw, tb, pw, pb, gw, gb, Pt, Pp, Pg);

    // NB * 72 chunks = 144 waves
    smat_kernel<<<(NB * 72) / 8, 256, 0, stream>>>(Pp, Pg, S);

    fold_kernel<<<(NB * NC * NCI + 255) / 256, 256, 0, stream>>>(S, rw, M64);

    // NB * 576 col strips = 1152 waves
    out_kernel<<<(NB * (NPIX / 16)) / 8, 256, 0, stream>>>(
        x, Pt, M64, rb, (float*)d_out);
}